// MichaelsRNN_58574763983457
// MI455X (gfx1250) — compile-verified
//
#include <hip/hip_runtime.h>

typedef float v2f __attribute__((ext_vector_type(2)));
typedef float v8f __attribute__((ext_vector_type(8)));

#define NN      300      // neurons
#define NFEAT   20       // input features
#define OUTD    50       // readout dim
#define NPMD    100      // readout input dim
#define BATCH   1024
#define TSTEPS  500
#define BT      16       // batch rows per workgroup
#define NTILES  19       // ceil(300/16)
#define KSTEPS  75       // 300/4
#define RSTRIDE 308      // LDS row stride (floats), even, spreads banks

__device__ __forceinline__ float retanh_f32(float x) {
#if __has_builtin(__builtin_amdgcn_tanh_f32)
    return fmaxf(__builtin_amdgcn_tanh_f32(x), 0.0f);   // v_tanh_f32 (TRANS32)
#elif __has_builtin(__builtin_amdgcn_tanhf)
    return fmaxf(__builtin_amdgcn_tanhf(x), 0.0f);
#else
    return fmaxf(tanhf(x), 0.0f);
#endif
}

// Pre-pack J into WMMA B-fragment layout:
// B[k,n] = J[n,k]; per (tile,kstep) block: 32 lanes x float2, lanes 0-15 hold
// (k0,k0+1) for n = tile*16+lane, lanes 16-31 hold (k0+2,k0+3). Zero-pad n>=300.
__global__ void pack_J_kernel(const float* __restrict__ J, float* __restrict__ Jp) {
    int blk  = blockIdx.x;          // tile*KSTEPS + kstep
    int lane = threadIdx.x;         // 0..31
    int tile  = blk / KSTEPS;
    int kstep = blk % KSTEPS;
    int n  = tile * 16 + (lane & 15);
    int kk = kstep * 4 + (lane >> 4) * 2;
    v2f val = {0.0f, 0.0f};
    if (n < NN) {
        val.x = J[n * NN + kk];
        val.y = J[n * NN + kk + 1];
    }
    ((v2f*)Jp)[(size_t)blk * 32 + lane] = val;
}

__global__ __launch_bounds__(NTILES * 32) void michaels_rnn_kernel(
    const float* __restrict__ data,   // (T, NF+1, B)
    const float* __restrict__ Jp,     // packed J fragments
    const float* __restrict__ I,      // (NN, NF)
    const float* __restrict__ S,      // (NN, 1)
    const float* __restrict__ Bb,     // (NN, 1)
    const float* __restrict__ x0,     // (NN,)
    const float* __restrict__ fc_w,   // (OUT, NPM)
    const float* __restrict__ fc_b,   // (OUT,)
    float* __restrict__ out)          // (T, B, OUT)
{
    __shared__ float r_lds[BT * RSTRIDE];

    const int lane  = threadIdx.x & 31;
    const int wave  = threadIdx.x >> 5;   // 0..18 == recur N-tile index
    const int hi    = lane >> 4;          // 0/1: selects K pair / M half
    const int lo    = lane & 15;
    const int bbase = blockIdx.x * BT;

    const int  n      = wave * 16 + lo;   // this lane's output neuron column
    const bool nvalid = (n < NN);

    // Per-lane constants for the epilogue (hoisted: S, Bb are time-invariant)
    const float sn = nvalid ? S[n]  : 0.0f;
    const float bn = nvalid ? Bb[n] : 0.0f;

    // Input-weight B-fragments (K = 20 -> 5 K-steps), kept in registers.
    v2f ifrag[5];
#pragma unroll
    for (int k5 = 0; k5 < 5; ++k5) {
        int kk = k5 * 4 + hi * 2;
        v2f v = {0.0f, 0.0f};
        if (nvalid) {
            v.x = I[n * NFEAT + kk];
            v.y = I[n * NFEAT + kk + 1];
        }
        ifrag[k5] = v;
    }

    // Readout-weight B-fragments (waves 0..3 only; K = 100 -> 25 K-steps).
    const int nout = wave * 16 + lo;      // readout column for waves 0..3
    v2f wfrag[25];
#pragma unroll
    for (int k25 = 0; k25 < 25; ++k25) {
        v2f v = {0.0f, 0.0f};
        if (wave < 4 && nout < OUTD) {
            int kk = k25 * 4 + hi * 2;
            v.x = fc_w[nout * NPMD + kk];
            v.y = fc_w[nout * NPMD + kk + 1];
        }
        wfrag[k25] = v;
    }
    const float fcb = (wave < 4 && nout < OUTD) ? fc_b[nout] : 0.0f;

    // State init: x[m,n] = x0[n] for all batch rows; r = retanh(x).
    const float xinit = nvalid ? x0[n] : 0.0f;
    float xv[8];
#pragma unroll
    for (int v = 0; v < 8; ++v) xv[v] = xinit;
    const float rinit = retanh_f32(xinit);
#pragma unroll
    for (int v = 0; v < 8; ++v) {
        int m = v + 8 * hi;
        r_lds[m * RSTRIDE + n] = rinit;
    }
    __syncthreads();

    for (int t = 0; t < TSTEPS; ++t) {
        const float* dt = data + (size_t)t * (NFEAT + 1) * BATCH;

        // Opaque barrier on the *offset* (not the pointer): LICM cannot hoist
        // the 75 J-fragment loads out of the t-loop (which would need ~150
        // VGPRs and spill to scratch), while address-space inference still
        // flows from Jp -> pure global_load (LOADcnt only, no FLAT/DScnt
        // coupling). J stays streamed from L2 each step.
        size_t joff = (size_t)wave * KSTEPS * 64;
        asm volatile("" : "+v"(joff));
        const float* Jt = Jp + joff;

        // ---- recur: two independent fp32 WMMA chains, branch-free pairs ----
        v8f acc0 = {};
        v8f acc1 = {};
#pragma unroll
        for (int ks = 0; ks < KSTEPS - 1; ks += 2) {     // 74 even ksteps
            int kb0 = ks * 4 + hi * 2;
            v2f a0 = *(const v2f*)&r_lds[lo * RSTRIDE + kb0];      // A[m,k] = r[m,k]
            v2f b0 = ((const v2f*)Jt)[ks * 32 + lane];             // B[k,n] = J[n,k]
            acc0 = __builtin_amdgcn_wmma_f32_16x16x4_f32(
                false, a0, false, b0, (short)0, acc0, false, false);

            int kb1 = kb0 + 4;
            v2f a1 = *(const v2f*)&r_lds[lo * RSTRIDE + kb1];
            v2f b1 = ((const v2f*)Jt)[(ks + 1) * 32 + lane];
            acc1 = __builtin_amdgcn_wmma_f32_16x16x4_f32(
                false, a1, false, b1, (short)0, acc1, false, false);
        }
        {   // peeled kstep 74
            int kb = (KSTEPS - 1) * 4 + hi * 2;
            v2f a = *(const v2f*)&r_lds[lo * RSTRIDE + kb];
            v2f b = ((const v2f*)Jt)[(KSTEPS - 1) * 32 + lane];
            acc0 = __builtin_amdgcn_wmma_f32_16x16x4_f32(
                false, a, false, b, (short)0, acc0, false, false);
        }
        // ---- inp: 5 ksteps from data, alternating chains ----
#pragma unroll
        for (int k5 = 0; k5 < 5; ++k5) {
            int kb = k5 * 4 + hi * 2;
            v2f a;
            a.x = dt[(size_t)kb * BATCH + bbase + lo];             // A[m,k] = image[k, b]
            a.y = dt[(size_t)(kb + 1) * BATCH + bbase + lo];
            if (k5 & 1) {
                acc1 = __builtin_amdgcn_wmma_f32_16x16x4_f32(
                    false, a, false, ifrag[k5], (short)0, acc1, false, false);
            } else {
                acc0 = __builtin_amdgcn_wmma_f32_16x16x4_f32(
                    false, a, false, ifrag[k5], (short)0, acc0, false, false);
            }
        }
        __syncthreads();   // everyone done reading r_t

        // ---- elementwise update: pre = x + (-x + acc + hold*S + Bb)/10 ----
        const float* holdt = dt + (size_t)NFEAT * BATCH + bbase;
        float rout[8];
#pragma unroll
        for (int v = 0; v < 8; ++v) {
            int m = v + 8 * hi;
            float hold = holdt[m];
            float accv = acc0[v] + acc1[v];
            float pre  = xv[v] + (accv - xv[v] + hold * sn + bn) * 0.1f;
            xv[v]   = pre;
            rout[v] = retanh_f32(pre);
        }
#pragma unroll
        for (int v = 0; v < 8; ++v) {
            int m = v + 8 * hi;
            r_lds[m * RSTRIDE + n] = rout[v];
        }
        __syncthreads();   // r_{t+1} visible

        // ---- readout: y = out[:, :100] @ fc_w.T + fc_b (waves 0..3) ----
        if (wave < 4) {
            v8f yacc0 = {};
            v8f yacc1 = {};
#pragma unroll
            for (int ks = 0; ks < 25; ++ks) {
                int kb = ks * 4 + hi * 2;
                v2f a = *(const v2f*)&r_lds[lo * RSTRIDE + kb];
                if (ks & 1) {
                    yacc1 = __builtin_amdgcn_wmma_f32_16x16x4_f32(
                        false, a, false, wfrag[ks], (short)0, yacc1, false, false);
                } else {
                    yacc0 = __builtin_amdgcn_wmma_f32_16x16x4_f32(
                        false, a, false, wfrag[ks], (short)0, yacc0, false, false);
                }
            }
            if (nout < OUTD) {
                float* outp = out + (size_t)t * BATCH * OUTD
                                  + (size_t)bbase * OUTD + nout;
#pragma unroll
                for (int v = 0; v < 8; ++v) {
                    int m = v + 8 * hi;
                    outp[(size_t)m * OUTD] = yacc0[v] + yacc1[v] + fcb;
                }
            }
        }
    }
}

extern "C" void kernel_launch(void* const* d_in, const int* in_sizes, int n_in,
                              void* d_out, int out_size, void* d_ws, size_t ws_size,
                              hipStream_t stream) {
    const float* data = (const float*)d_in[0];
    const float* J    = (const float*)d_in[1];
    const float* I    = (const float*)d_in[2];
    const float* S    = (const float*)d_in[3];
    const float* Bb   = (const float*)d_in[4];
    const float* x0   = (const float*)d_in[5];
    const float* fc_w = (const float*)d_in[6];
    const float* fc_b = (const float*)d_in[7];
    float* out = (float*)d_out;

    float* Jp = (float*)d_ws;  // NTILES*KSTEPS*64 floats = 364,800 bytes

    pack_J_kernel<<<NTILES * KSTEPS, 32, 0, stream>>>(J, Jp);
    michaels_rnn_kernel<<<BATCH / BT, NTILES * 32, 0, stream>>>(
        data, Jp, I, S, Bb, x0, fc_w, fc_b, out);
}